// TrainNhpp3Nm_6098853560634
// MI455X (gfx1250) — compile-verified
//
#include <hip/hip_runtime.h>
#include <hip/hip_bf16.h>

#define L2E  1.44269504088896340736f   // log2(e)
#define LN2  0.69314718055994530942f

// ---------------------------------------------------------------------------
// Kernel 1: fold everything that does not depend on the event index into
// 6 scalar constants living in d_ws:
//   ws[0..4] = quartic coefficients c0..c4 of |dx + dv t + 0.5 da t^2|^2
//   ws[5]    = (gamma0 + gamma1) * log2(e)
// ---------------------------------------------------------------------------
__global__ void nhpp_precompute(const float* __restrict__ gamma,
                                const float* __restrict__ x0,
                                const float* __restrict__ v0,
                                const float* __restrict__ a0,
                                float* __restrict__ ws) {
    if (threadIdx.x == 0 && blockIdx.x == 0) {
        float c0 = 0.f, c1 = 0.f, c2 = 0.f, c3 = 0.f, c4 = 0.f;
#pragma unroll
        for (int d = 0; d < 3; ++d) {
            float dx = x0[d] - x0[3 + d];
            float dv = v0[d] - v0[3 + d];
            float da = a0[d] - a0[3 + d];
            c0 = fmaf(dx, dx, c0);
            c1 = fmaf(2.0f * dx, dv, c1);
            c2 = fmaf(dv, dv, fmaf(dx, da, c2));
            c3 = fmaf(dv, da, c3);
            c4 = fmaf(0.25f * da, da, c4);
        }
        ws[0] = c0; ws[1] = c1; ws[2] = c2; ws[3] = c3; ws[4] = c4;
        ws[5] = (gamma[0] + gamma[1]) * L2E;
    }
}

// Per-event contribution: ci - 1{k>0} * log(ci)
__device__ __forceinline__ float event_ll(float t0, float t1, int kk,
                                          float c0, float c1, float c2,
                                          float c3, float c4, float gl2e) {
    const float step = (t1 - t0) * 0.1f;
    float t = t0;
    float s = 0.f;
#pragma unroll
    for (int n = 0; n < 10; ++n) {
        // q(t) = c0 + t(c1 + t(c2 + t(c3 + t*c4)))  -- 4 FMAs
        float q = fmaf(t, fmaf(t, fmaf(t, fmaf(t, c4, c3), c2), c1), c0);
        float dist = __builtin_amdgcn_sqrtf(q);                 // v_sqrt_f32
        s += __builtin_amdgcn_exp2f(fmaf(dist, -L2E, gl2e));    // v_exp_f32
        t += step;
    }
    float ci = s * step;
    float lg = __builtin_amdgcn_logf(ci) * LN2;                 // v_log_f32
    return (kk > 0) ? (ci - lg) : ci;
}

// ---------------------------------------------------------------------------
// Kernel 2: grid-stride over events, 4 events per thread via 128-bit loads,
// wave32 shuffle reduce -> LDS block reduce -> one f32 atomic per block.
// ---------------------------------------------------------------------------
__global__ __launch_bounds__(256) void nhpp_main(const float* __restrict__ tInit,
                                                 const float* __restrict__ tLast,
                                                 const int*   __restrict__ kArr,
                                                 const float* __restrict__ ws,
                                                 float* __restrict__ out, int E) {
    const float c0 = ws[0], c1 = ws[1], c2 = ws[2], c3 = ws[3], c4 = ws[4];
    const float gl2e = ws[5];

    const int nthreads = (int)(gridDim.x * blockDim.x);
    const int tid      = (int)(blockIdx.x * blockDim.x + threadIdx.x);
    const int nvec     = E >> 2;

    const float4* __restrict__ t0v = (const float4*)tInit;
    const float4* __restrict__ t1v = (const float4*)tLast;
    const int4*   __restrict__ kv  = (const int4*)kArr;

    float acc = 0.f;
    for (int i = tid; i < nvec; i += nthreads) {
        const int j = i + nthreads;                 // prefetch next stride chunk
        if (j < nvec) {
            __builtin_prefetch(&t0v[j], 0, 3);      // global_prefetch_b8
            __builtin_prefetch(&t1v[j], 0, 3);
            __builtin_prefetch(&kv[j],  0, 3);
        }
        float4 a  = t0v[i];                         // global_load_b128
        float4 b  = t1v[i];
        int4   kk = kv[i];
        acc += event_ll(a.x, b.x, kk.x, c0, c1, c2, c3, c4, gl2e);
        acc += event_ll(a.y, b.y, kk.y, c0, c1, c2, c3, c4, gl2e);
        acc += event_ll(a.z, b.z, kk.z, c0, c1, c2, c3, c4, gl2e);
        acc += event_ll(a.w, b.w, kk.w, c0, c1, c2, c3, c4, gl2e);
    }
    // scalar tail (E not multiple of 4)
    for (int i = (nvec << 2) + tid; i < E; i += nthreads)
        acc += event_ll(tInit[i], tLast[i], kArr[i], c0, c1, c2, c3, c4, gl2e);

    // ---- wave32 reduction ----
#pragma unroll
    for (int off = 16; off > 0; off >>= 1)
        acc += __shfl_down(acc, (unsigned)off, 32);

    __shared__ float smem[8];
    const int lane = (int)(threadIdx.x & 31u);
    const int wid  = (int)(threadIdx.x >> 5u);
    if (lane == 0) smem[wid] = acc;
    __syncthreads();
    if (wid == 0) {
        float v = (lane < 8) ? smem[lane] : 0.f;
#pragma unroll
        for (int off = 4; off > 0; off >>= 1)
            v += __shfl_down(v, (unsigned)off, 32);
        if (lane == 0)
            unsafeAtomicAdd(out, v);                // global_atomic_add_f32
    }
}

extern "C" void kernel_launch(void* const* d_in, const int* in_sizes, int n_in,
                              void* d_out, int out_size, void* d_ws, size_t ws_size,
                              hipStream_t stream) {
    const float* tInit = (const float*)d_in[0];
    const float* tLast = (const float*)d_in[1];
    const int*   kArr  = (const int*)d_in[2];
    const float* gamma = (const float*)d_in[3];
    const float* x0    = (const float*)d_in[4];
    const float* v0    = (const float*)d_in[5];
    const float* a0    = (const float*)d_in[6];
    float*       out   = (float*)d_out;
    float*       ws    = (float*)d_ws;

    const int E = in_sizes[0];

    // capture-safe zero of the scalar accumulator
    hipMemsetAsync(d_out, 0, (size_t)out_size * sizeof(float), stream);

    nhpp_precompute<<<1, 32, 0, stream>>>(gamma, x0, v0, a0, ws);

    const int block = 256;
    int nvec = E >> 2;
    int grid = (nvec + block - 1) / block;
    if (grid < 1) grid = 1;
    if (grid > 65535) grid = 65535;
    nhpp_main<<<grid, block, 0, stream>>>(tInit, tLast, kArr, ws, out, E);
}